// Group_90623809946178
// MI455X (gfx1250) — compile-verified
//
#include <hip/hip_runtime.h>
#include <hip/hip_bf16.h>

#define NUM_GROUP 2048
#define GROUP_SIZE 16
#define NPTS 16384
#define BATCH 4

typedef __attribute__((ext_vector_type(2))) float v2f;
typedef __attribute__((ext_vector_type(8))) float v8f;

// ------------------------------------------------------------------
// Kernel 1: Furthest Point Sampling. One workgroup per batch.
// 1024 threads x 16 points each; points AND running min-distances
// live in VGPRs for the whole 2047-step sequential chain.
// ------------------------------------------------------------------
#define FPS_THREADS 1024
#define PPT (NPTS / FPS_THREADS) // 16

__global__ __launch_bounds__(FPS_THREADS)
void fps_kernel(const float* __restrict__ pcd, float* __restrict__ centers)
{
    const int b = blockIdx.x;
    const int t = threadIdx.x;
    const float* xyz  = pcd + (size_t)b * NPTS * 3;
    float*       cent = centers + (size_t)b * NUM_GROUP * 3;

    __shared__ float s_last[3];
    __shared__ float s_val[32];
    __shared__ int   s_idx[32];

    float px[PPT], py[PPT], pz[PPT], md[PPT];
    #pragma unroll
    for (int i = 0; i < PPT; ++i) {
        const int j = t + (i << 10);
        px[i] = xyz[3*j+0];
        py[i] = xyz[3*j+1];
        pz[i] = xyz[3*j+2];
        md[i] = 1e10f;
    }

    if (t == 0) {
        s_last[0] = xyz[0]; s_last[1] = xyz[1]; s_last[2] = xyz[2];
        cent[0] = xyz[0]; cent[1] = xyz[1]; cent[2] = xyz[2];
    }
    __syncthreads();

    const int lane = t & 31;
    const int wave = t >> 5;

    for (int g = 1; g < NUM_GROUP; ++g) {
        const float bx = s_last[0], by = s_last[1], bz = s_last[2];
        float bestv = -1.0f;
        int   besti = NPTS;
        #pragma unroll
        for (int i = 0; i < PPT; ++i) {
            const float dx = px[i]-bx, dy = py[i]-by, dz = pz[i]-bz;
            const float d  = dx*dx + dy*dy + dz*dz;
            const float m  = fminf(md[i], d);
            md[i] = m;
            const int j = t + (i << 10);
            const bool better = (m > bestv) || ((m == bestv) && (j < besti));
            bestv = better ? m : bestv;
            besti = better ? j : besti;
        }
        // wave32 argmax (tie -> lowest index, matching jnp.argmax)
        #pragma unroll
        for (int off = 16; off >= 1; off >>= 1) {
            const float ov = __shfl_down(bestv, off, 32);
            const int   oi = __shfl_down(besti, off, 32);
            const bool better = (ov > bestv) || ((ov == bestv) && (oi < besti));
            bestv = better ? ov : bestv;
            besti = better ? oi : besti;
        }
        if (lane == 0) { s_val[wave] = bestv; s_idx[wave] = besti; }
        __syncthreads();
        if (wave == 0) {
            bestv = s_val[lane]; besti = s_idx[lane];
            #pragma unroll
            for (int off = 16; off >= 1; off >>= 1) {
                const float ov = __shfl_down(bestv, off, 32);
                const int   oi = __shfl_down(besti, off, 32);
                const bool better = (ov > bestv) || ((ov == bestv) && (oi < besti));
                bestv = better ? ov : bestv;
                besti = better ? oi : besti;
            }
            if (lane == 0) {
                const float wx = xyz[3*besti+0];
                const float wy = xyz[3*besti+1];
                const float wz = xyz[3*besti+2];
                s_last[0] = wx; s_last[1] = wy; s_last[2] = wz;
                cent[3*g+0] = wx; cent[3*g+1] = wy; cent[3*g+2] = wz;
            }
        }
        __syncthreads();
    }
}

// ------------------------------------------------------------------
// Kernel 2: kNN (top-16 nearest) + neighborhood gather.
// One wave per tile of 16 centers. Dot products c.x via
// V_WMMA_F32_16X16X4_F32 (K=3 padded to 4); ranking key
// d' = |x|^2 - 2 c.x (row-constant |c|^2 dropped — same ordering).
// Lanes 0..15 each own one center and keep a sorted top-16
// (dist,idx) list in registers; stable strict-< insertion matches
// lax.top_k nearest-first / lowest-index-on-tie ordering.
// ------------------------------------------------------------------
__global__ __launch_bounds__(32)
void knn_group_kernel(const float* __restrict__ pcd,
                      const float* __restrict__ centers,
                      float* __restrict__ nbrOut)
{
    const int tileId = blockIdx.x;              // 0 .. B*G/16-1
    const int b  = tileId / (NUM_GROUP / 16);
    const int tg = tileId % (NUM_GROUP / 16);
    const int g0 = tg * 16;
    const int lane = threadIdx.x;

    const float* xyz  = pcd + (size_t)b * NPTS * 3;
    const float* cent = centers + (size_t)b * NUM_GROUP * 3;

    __shared__ float s_dot[16 * 16];
    __shared__ float s_ptsq[16];

    // A matrix (16x4 f32): lanes 0-15 -> (K0,K1)=(x,y), lanes 16-31 -> (K2,K3)=(z,0)
    const int cm = lane & 15;
    const float* cp = cent + 3 * (g0 + cm);
    v2f a;
    if (lane < 16) { a.x = cp[0]; a.y = cp[1]; }
    else           { a.x = cp[2]; a.y = 0.0f;  }

    float kd[GROUP_SIZE];
    int   ki[GROUP_SIZE];
    #pragma unroll
    for (int m = 0; m < GROUP_SIZE; ++m) { kd[m] = 1e30f; ki[m] = 0; }

    for (int base = 0; base < NPTS; base += 16) {
        const int pcol = lane & 15;
        const float* pp = xyz + 3 * (base + pcol);
        const float x = pp[0], y = pp[1], z = pp[2];
        v2f bm;
        if (lane < 16) {
            bm.x = x; bm.y = y;
            s_ptsq[pcol] = x*x + y*y + z*z;
        } else {
            bm.x = z; bm.y = 0.0f;
        }
        __syncthreads();

        v8f acc = {};
        // 8 args: (neg_a, A, neg_b, B, c_mod, C, reuse_a, reuse_b)
        acc = __builtin_amdgcn_wmma_f32_16x16x4_f32(false, a, false, bm,
                                                    (short)0, acc, false, false);

        // D layout: lanes 0-15: vgpr v -> row v; lanes 16-31: row v+8; col = lane&15
        const int row0 = (lane < 16) ? 0 : 8;
        #pragma unroll
        for (int v = 0; v < 8; ++v)
            s_dot[(row0 + v) * 16 + (lane & 15)] = acc[v];
        __syncthreads();

        if (lane < 16) {
            const int row = lane;
            for (int c = 0; c < 16; ++c) {
                const float d = s_ptsq[c] - 2.0f * s_dot[row * 16 + c];
                if (d < kd[GROUP_SIZE - 1]) {
                    float cd = d; int ci = base + c;
                    #pragma unroll
                    for (int m = 0; m < GROUP_SIZE; ++m) {
                        const bool sw = cd < kd[m];
                        const float td = kd[m]; const int ti = ki[m];
                        kd[m] = sw ? cd : kd[m];
                        ki[m] = sw ? ci : ki[m];
                        cd = sw ? td : cd;
                        ci = sw ? ti : ci;
                    }
                }
            }
        }
        __syncthreads();
    }

    // neighborhood = xyz[idx] - center
    if (lane < 16) {
        const int g = g0 + lane;
        const float* cc = cent + 3 * g;
        const float cx = cc[0], cy = cc[1], cz = cc[2];
        float* out = nbrOut + ((size_t)(b * NUM_GROUP + g)) * GROUP_SIZE * 3;
        #pragma unroll
        for (int m = 0; m < GROUP_SIZE; ++m) {
            const float* p = xyz + 3 * ki[m];
            out[3*m+0] = p[0] - cx;
            out[3*m+1] = p[1] - cy;
            out[3*m+2] = p[2] - cz;
        }
    }
}

extern "C" void kernel_launch(void* const* d_in, const int* in_sizes, int n_in,
                              void* d_out, int out_size, void* d_ws, size_t ws_size,
                              hipStream_t stream)
{
    (void)in_sizes; (void)n_in; (void)d_ws; (void)ws_size; (void)out_size;

    const float* pcd = (const float*)d_in[0];
    float* out  = (float*)d_out;
    float* nbr  = out;                                                    // B,G,16,3
    float* cent = out + (size_t)BATCH * NUM_GROUP * GROUP_SIZE * 3;       // B,G,3

    fps_kernel<<<BATCH, FPS_THREADS, 0, stream>>>(pcd, cent);
    knn_group_kernel<<<BATCH * (NUM_GROUP / 16), 32, 0, stream>>>(pcd, cent, nbr);
}